// PlenoxelModel_41455024341760
// MI455X (gfx1250) — compile-verified
//
#include <hip/hip_runtime.h>
#include <hip/hip_bf16.h>
#include <math.h>

// Plenoxel render for MI455X (gfx1250, wave32).
// One wave per ray; lane = voxel channel for perfectly coalesced 112B corner
// gathers; SH contraction done as 7x V_WMMA_F32_16X16X4_F32 (K=28) in f32.
// Streamed inputs (positions/deltas) loaded non-temporal so the 192MB L2 is
// dedicated to the 235MB voxel grid (~32x average reuse per grid byte).

typedef __attribute__((ext_vector_type(2))) float v2f;
typedef __attribute__((ext_vector_type(8))) float v8f;

#define GDIM 128
#define VDIM 28
#define NRAY 65536
#define NSAMP 128
#define OFFZ (VDIM)               //    28 elements
#define OFFY (GDIM * VDIM)        //  3584 elements
#define OFFX (GDIM * GDIM * VDIM) // 458752 elements (byte offset fits imm24)

// Broadcast from a compile-time lane: v_readlane_b32 (no LDS traffic).
__device__ __forceinline__ float bcast(float x, int srcLane) {
    return __int_as_float(__builtin_amdgcn_readlane(__float_as_int(x), srcLane));
}

__global__ __launch_bounds__(256) void plenoxel_render(
    const float* __restrict__ grid,   // [128,128,128,28]
    const float* __restrict__ pos,    // [R,S,3]
    const float* __restrict__ del,    // [R,S]
    const float* __restrict__ vang,   // [2]
    float* __restrict__ out)          // [R,3]
{
    __shared__ float Ysh[9];
    __shared__ float Bsh[VDIM * 16];      // B matrix: 28 rows x 16 cols
    __shared__ float Ash[8][16 * VDIM];   // per-wave staging: 16 samples x 28 ch

    const int tid  = threadIdx.x;
    const int lane = tid & 31;
    const int wave = tid >> 5;
    const int ray  = blockIdx.x * 8 + wave;

    // ---- SH basis (one thread; identical for whole dispatch) ----
    if (tid == 0) {
        float th = vang[0], ph = vang[1];
        float st = __sinf(th), ct = __cosf(th);
        float sp = __sinf(ph), cp = __cosf(ph);
        const float Y00  = 0.28209479177387814f;  // 0.5*sqrt(1/pi)
        const float HS3  = 0.4886025119029199f;   // 0.5*sqrt(3/pi)
        const float QS5  = 0.31539156525252005f;  // 0.25*sqrt(5/pi)
        const float HS15 = 1.0925484305920792f;   // 0.5*sqrt(15/pi)
        const float QS15 = 0.5462742152960396f;   // 0.25*sqrt(15/pi)
        Ysh[0] = Y00;
        Ysh[1] = HS3 * st * sp;
        Ysh[2] = HS3 * ct;
        Ysh[3] = HS3 * st * cp;
        Ysh[4] = HS15 * st * cp * st * sp;
        Ysh[5] = HS15 * st * sp * ct;
        Ysh[6] = QS5 * (3.0f * ct * ct - 1.0f);
        Ysh[7] = HS15 * st * cp * ct;
        Ysh[8] = QS15 * ((st * cp) * (st * cp) - (st * sp) * (st * sp));
    }
    __syncthreads();

    // ---- Build B (28x16): row 0 -> sigma into col 3; row 1+9c+k -> Y[k] into col c
    for (int idx = tid; idx < VDIM * 16; idx += 256) {
        int row = idx >> 4, n = idx & 15;
        float v = 0.0f;
        if (row == 0) {
            v = (n == 3) ? 1.0f : 0.0f;
        } else {
            int c = (row - 1) / 9, k = (row - 1) % 9;
            v = (n == c) ? Ysh[k] : 0.0f;
        }
        Bsh[idx] = v;
    }
    __syncthreads();

    // ---- Load B into WMMA layout: VGPR0 lanes0-15 = K=4j, lanes16-31 = K=4j+2;
    //      VGPR1 = K=4j+1 / 4j+3. N = lane%16.
    const int nn = lane & 15;
    const int hf = lane >> 4;
    v2f bmat[7];
#pragma unroll
    for (int j = 0; j < 7; ++j) {
        int r0 = 4 * j + 2 * hf;
        v2f b;
        b.x = Bsh[r0 * 16 + nn];
        b.y = Bsh[(r0 + 1) * 16 + nn];
        bmat[j] = b;
    }

    float* As = &Ash[wave][0];
    const int aoff = nn * VDIM + hf * 2;            // A-matrix per-lane base (elements)
    const int chan = (lane < VDIM) ? lane : 0;      // lanes 28..31 harmless dup loads

    const float* prow = pos + (size_t)ray * NSAMP * 3;
    const float* drow = del + (size_t)ray * NSAMP;

    float T = 0.0f, accR = 0.0f, accG = 0.0f, accB = 0.0f;

    for (int s0 = 0; s0 < NSAMP; s0 += 16) {
        // ---- Gather + trilinear: 8 coalesced 112B corner loads per sample ----
        for (int t = 0; t < 16; ++t) {
            const float* pp = prow + (size_t)(s0 + t) * 3;
            float px = __builtin_nontemporal_load(pp + 0);   // stream, keep L2 for grid
            float py = __builtin_nontemporal_load(pp + 1);
            float pz = __builtin_nontemporal_load(pp + 2);
            float fx = floorf(px), fy = floorf(py), fz = floorf(pz);
            float dx = px - fx, dy = py - fy, dz = pz - fz;
            int ix = (int)fx, iy = (int)fy, iz = (int)fz;
            // positions in [0,126) => all 8 corners in-bounds, no clamping needed
            const float* g = grid + (((ix * GDIM + iy) * GDIM + iz) * VDIM + chan);
            float w00 = (1.0f - dx) * (1.0f - dy);
            float w01 = (1.0f - dx) * dy;
            float w10 = dx * (1.0f - dy);
            float w11 = dx * dy;
            float z0 = 1.0f - dz, z1 = dz;
            float acc;
            acc  =      g[0]                  * (w00 * z0);
            acc  = fmaf(g[OFFZ],                w00 * z1, acc);
            acc  = fmaf(g[OFFY],                w01 * z0, acc);
            acc  = fmaf(g[OFFY + OFFZ],         w01 * z1, acc);
            acc  = fmaf(g[OFFX],                w10 * z0, acc);
            acc  = fmaf(g[OFFX + OFFZ],         w10 * z1, acc);
            acc  = fmaf(g[OFFX + OFFY],         w11 * z0, acc);
            acc  = fmaf(g[OFFX + OFFY + OFFZ],  w11 * z1, acc);
            if (lane < VDIM) As[t * VDIM + lane] = acc;   // stage sample-major
        }

        // ---- D(16x16) = A(16x28) x B(28x16) in f32: cols 0..2 = RGB, col 3 = sigma
        v8f cacc = {};
#pragma unroll
        for (int j = 0; j < 7; ++j) {
            v2f a = *(const v2f*)(As + aoff + 4 * j);   // ds_load_b64, 8B aligned
            cacc = __builtin_amdgcn_wmma_f32_16x16x4_f32(
                false, a, false, bmat[j], (short)0, cacc, false, false);
        }

        // ---- Sequential transmittance scan (T = cumsum incl. current sample) ----
#pragma unroll
        for (int m = 0; m < 16; ++m) {
            const int vi = m & 7;            // D VGPR holding sample m
            const int bl = (m >> 3) * 16;    // lane half holding sample m
            float Rv = bcast(cacc[vi], bl + 0);
            float Gv = bcast(cacc[vi], bl + 1);
            float Bv = bcast(cacc[vi], bl + 2);
            float sg = bcast(cacc[vi], bl + 3);
            float dlt = __builtin_nontemporal_load(drow + s0 + m);  // streamed
            float att = __expf(-sg * dlt);
            T += att;
            float w = T * (1.0f - att);
            accR = fmaf(w, Rv, accR);
            accG = fmaf(w, Gv, accG);
            accB = fmaf(w, Bv, accB);
        }
    }

    float v = (lane == 0) ? accR : ((lane == 1) ? accG : accB);
    if (lane < 3) out[(size_t)ray * 3 + lane] = v;
}

extern "C" void kernel_launch(void* const* d_in, const int* in_sizes, int n_in,
                              void* d_out, int out_size, void* d_ws, size_t ws_size,
                              hipStream_t stream) {
    const float* grid = (const float*)d_in[0];   // voxel_grid [128,128,128,28]
    const float* pos  = (const float*)d_in[1];   // positions  [65536,128,3]
    const float* del  = (const float*)d_in[2];   // deltas     [65536,128]
    const float* vang = (const float*)d_in[3];   // viewing_angle [2]
    float* out = (float*)d_out;                  // [65536,3]

    dim3 block(256);                 // 8 wave32s = 8 rays per block
    dim3 gridd(NRAY / 8);            // 8192 blocks
    plenoxel_render<<<gridd, block, 0, stream>>>(grid, pos, del, vang, out);
}